// MLKA_13237089206460
// MI455X (gfx1250) — compile-verified
//
#include <hip/hip_runtime.h>

typedef __attribute__((ext_vector_type(2))) float v2f;
typedef __attribute__((ext_vector_type(8))) float v8f;

#define SP 131072   // D*H*W = 32*64*64 spatial elements per channel
#define NB 2        // batch
#define HW 4096     // H*W

static __device__ __forceinline__ v8f wmma_f32_4(v2f a, v2f b, v8f c) {
    return __builtin_amdgcn_wmma_f32_16x16x4_f32(false, a, false, b, (short)0, c, false, false);
}

// Load a rows x 16 tile from global (channel stride SP) into LDS [rows][16], float4 per lane.
static __device__ __forceinline__ void load_tile(float* dst, const float* src, int rows, int lane) {
    const int iters = rows * 16 / 128;
    for (int j = 0; j < iters; ++j) {
        int e  = (j * 32 + lane) * 4;
        int ch = e >> 4, no = e & 15;
        *reinterpret_cast<float4*>(&dst[ch * 16 + no]) =
            *reinterpret_cast<const float4*>(&src[(size_t)ch * SP + no]);
    }
}

// ---------------------------------------------------------------------------
// Kernel 1: fused channel-LayerNorm + 1x1x1 conv 64->128 (WMMA f32 16x16x4)
// Each wave: 16 consecutive spatial positions, all 128 output channels.
// ---------------------------------------------------------------------------
__global__ __launch_bounds__(128) void pf_ln_wmma(
    const float* __restrict__ x, const float* __restrict__ lnw_g,
    const float* __restrict__ lnb_g, const float* __restrict__ wpf_g,
    const float* __restrict__ bpf_g, float* __restrict__ y)
{
    __shared__ float wl[128 * 64];          // 32 KB: w_pf row-major (oc, ic)
    __shared__ float lnw[64], lnb[64], bpf[128];
    __shared__ __align__(16) float tiles[4][1056];   // per wave: 64x16 tile + 32 stats

    int tid = threadIdx.x;
    for (int i = tid; i < 8192; i += 128) wl[i] = wpf_g[i];
    if (tid < 64)  { lnw[tid] = lnw_g[tid]; lnb[tid] = lnb_g[tid]; }
    if (tid < 128) bpf[tid] = bpf_g[tid];

    int wv = tid >> 5, lane = tid & 31;
    int g  = blockIdx.x * 4 + wv;           // 16384 groups total
    int b  = g >> 13;                       // 8192 groups per batch
    int s0 = (g & 8191) << 4;
    float* tile = tiles[wv];
    const float* xb = x + (size_t)b * 64 * SP + s0;

    load_tile(tile, xb, 64, lane);          // 64 channels x 16 positions
    __syncthreads();

    if (lane < 16) {                        // per-position LN stats over 64 channels
        float s = 0.f, q = 0.f;
        for (int c = 0; c < 64; ++c) { float v = tile[c * 16 + lane]; s += v; q += v * v; }
        float u   = s * (1.0f / 64.0f);
        float var = q * (1.0f / 64.0f) - u * u;
        tile[1024 + lane] = u;
        tile[1040 + lane] = rsqrtf(var + 1e-6f);
    }
    __syncthreads();

    int n = lane & 15, m = lane & 15;
    int khalf = (lane >> 4) << 1;           // 0 or 2 (A/B K placement per half-wave)
    float un = tile[1024 + n], rs = tile[1040 + n];

    v8f acc[8] = {};                        // 8 M-tiles of 16x16 (128 out channels)
    for (int kk = 0; kk < 64; kk += 4) {
        int k0 = kk + khalf;
        v2f bv;
        bv.x = (tile[k0 * 16 + n] - un) * rs * lnw[k0] + lnb[k0];
        bv.y = (tile[(k0 + 1) * 16 + n] - un) * rs * lnw[k0 + 1] + lnb[k0 + 1];
#pragma unroll
        for (int mt = 0; mt < 8; ++mt) {
            v2f av;
            av.x = wl[(mt * 16 + m) * 64 + k0];
            av.y = wl[(mt * 16 + m) * 64 + k0 + 1];
            acc[mt] = wmma_f32_4(av, bv, acc[mt]);
        }
    }

    int hi = (lane >> 4) * 8;
    float* yb = y + (size_t)b * 128 * SP + s0;
#pragma unroll
    for (int mt = 0; mt < 8; ++mt)
#pragma unroll
        for (int r = 0; r < 8; ++r) {
            int ch = mt * 16 + r + hi;
            yb[(size_t)ch * SP + n] = acc[mt][r] + bpf[ch];
        }
}

// ---------------------------------------------------------------------------
// Kernel 2: LDS-tiled depthwise 3D conv, templated on kernel size / dilation.
// One block per (b, ch, d, 8x32 hw-tile). Because the output tile spans a
// single d-plane, the dilated taps touch exactly KS input planes at stride
// DIL, so the staged tile is KS x (TH+2*HLO) x (TW+2*HLO) floats (<= 36 KB
// for the 7^3/dil-3 case). Halo zeros are materialized at load time, so the
// 343-tap inner loop is pure unrolled v_fma_f32 on LDS with no branches.
// ---------------------------------------------------------------------------
template <int KS, int DIL, int TH, int TW>
__global__ __launch_bounds__(256) void dw_conv_tiled(
    const float* __restrict__ in, int inChTot, int inCh0,
    const float* __restrict__ w, const float* __restrict__ bias,
    float* __restrict__ out)
{
    constexpr int R   = KS / 2;
    constexpr int HLO = DIL * R;           // halo in h and w
    constexpr int RH  = TH + 2 * HLO;      // staged rows per plane
    constexpr int RW  = TW + 2 * HLO;      // staged cols per row
    constexpr int K3  = KS * KS * KS;
    constexpr int TSZ = KS * RH * RW;      // staged elements
    constexpr int NHT = 64 / TH, NWT = 64 / TW;

    __shared__ float wl[K3];
    __shared__ float bsh;
    __shared__ float lt[TSZ];

    int bid = blockIdx.x;
    int wt = bid % NWT; bid /= NWT;
    int ht = bid % NHT; bid /= NHT;
    int d  = bid % 32;  bid /= 32;
    int ch = bid % 32;
    int b  = bid / 32;
    int h0 = ht * TH, w0 = wt * TW;
    int tid = threadIdx.x;

    for (int i = tid; i < K3; i += 256) wl[i] = w[ch * K3 + i];
    if (tid == 0) bsh = bias[ch];

    const float* ib = in + (size_t)(b * inChTot + inCh0 + ch) * SP;

    // stage KS dilated input planes (with zero halo) into LDS
    for (int i = tid; i < TSZ; i += 256) {
        int kd  = i / (RH * RW);
        int rem = i % (RH * RW);
        int row = rem / RW, col = rem % RW;
        int zd = d + DIL * (kd - R);
        int zh = h0 - HLO + row;
        int zw = w0 - HLO + col;
        float v = 0.f;
        if ((unsigned)zd < 32u && (unsigned)zh < 64u && (unsigned)zw < 64u)
            v = ib[(size_t)zd * HW + zh * 64 + zw];
        lt[i] = v;
    }
    __syncthreads();

    // one output per thread: LDS row = hr + DIL*kh, col = wr + DIL*kw
    int hr = tid / TW, wr = tid % TW;      // TH*TW == 256 == blockDim
    float acc = bsh;
#pragma unroll
    for (int kd = 0; kd < KS; ++kd) {
        const float* lp = &lt[kd * RH * RW];
#pragma unroll
        for (int kh = 0; kh < KS; ++kh) {
            const float* rowp = &lp[(hr + DIL * kh) * RW + wr];
            const float* wp   = &wl[(kd * KS + kh) * KS];
#pragma unroll
            for (int kw = 0; kw < KS; ++kw)
                acc = __builtin_fmaf(wp[kw], rowp[DIL * kw], acc);
        }
    }
    out[(size_t)(b * 32 + ch) * SP + (size_t)d * HW + (h0 + hr) * 64 + (w0 + wr)] = acc;
}

// ---------------------------------------------------------------------------
// Kernel 3: fused  (w3c GEMM)*g3 | (w5c GEMM)*g5 -> att ; xp*att ;
//           w_pl GEMM ; *scale + shortcut.  Three WMMA GEMMs per wave.
// ---------------------------------------------------------------------------
__global__ __launch_bounds__(128) void combine_wmma(
    const float* __restrict__ y,
    const float* __restrict__ u3, const float* __restrict__ g3,
    const float* __restrict__ u5, const float* __restrict__ g5,
    const float* __restrict__ w3c_g, const float* __restrict__ b3c_g,
    const float* __restrict__ w5c_g, const float* __restrict__ b5c_g,
    const float* __restrict__ wpl_g, const float* __restrict__ bpl_g,
    const float* __restrict__ scl_g, const float* __restrict__ x,
    float* __restrict__ out)
{
    __shared__ float w3c[1024], w5c[1024], wpl[4096];
    __shared__ float b3c[32], b5c[32], bpl[64], scl[64];
    __shared__ __align__(16) float tiles[4][1024];

    int tid = threadIdx.x;
    for (int i = tid; i < 1024; i += 128) { w3c[i] = w3c_g[i]; w5c[i] = w5c_g[i]; }
    for (int i = tid; i < 4096; i += 128) wpl[i] = wpl_g[i];
    if (tid < 32) { b3c[tid] = b3c_g[tid]; b5c[tid] = b5c_g[tid]; }
    if (tid < 64) { bpl[tid] = bpl_g[tid]; scl[tid] = scl_g[tid]; }

    int wv = tid >> 5, lane = tid & 31;
    int g  = blockIdx.x * 4 + wv;
    int b  = g >> 13;
    int s0 = (g & 8191) << 4;
    float* tile = tiles[wv];
    int n = lane & 15, m = lane & 15;
    int khalf = (lane >> 4) << 1;
    int hi = (lane >> 4) * 8;

    float att[4][8];   // att[mt][r] for out-channel ch = mt*16 + r + hi

    // ----- branch 3: lka3 = w3c @ u3 + b3c, gated by g3 -----
    load_tile(tile,       u3 + (size_t)b * 32 * SP + s0, 32, lane);
    load_tile(tile + 512, g3 + (size_t)b * 32 * SP + s0, 32, lane);
    __syncthreads();
    {
        v8f a3[2] = {};
        for (int kk = 0; kk < 32; kk += 4) {
            int k0 = kk + khalf;
            v2f bv; bv.x = tile[k0 * 16 + n]; bv.y = tile[(k0 + 1) * 16 + n];
#pragma unroll
            for (int mt = 0; mt < 2; ++mt) {
                v2f av;
                av.x = w3c[(mt * 16 + m) * 32 + k0];
                av.y = w3c[(mt * 16 + m) * 32 + k0 + 1];
                a3[mt] = wmma_f32_4(av, bv, a3[mt]);
            }
        }
#pragma unroll
        for (int mt = 0; mt < 2; ++mt)
#pragma unroll
            for (int r = 0; r < 8; ++r) {
                int ch = mt * 16 + r + hi;
                att[mt][r] = (a3[mt][r] + b3c[ch]) * tile[512 + ch * 16 + n];
            }
    }
    __syncthreads();

    // ----- branch 5: lka5 = w5c @ u5 + b5c, gated by g5 -----
    load_tile(tile,       u5 + (size_t)b * 32 * SP + s0, 32, lane);
    load_tile(tile + 512, g5 + (size_t)b * 32 * SP + s0, 32, lane);
    __syncthreads();
    {
        v8f a5[2] = {};
        for (int kk = 0; kk < 32; kk += 4) {
            int k0 = kk + khalf;
            v2f bv; bv.x = tile[k0 * 16 + n]; bv.y = tile[(k0 + 1) * 16 + n];
#pragma unroll
            for (int mt = 0; mt < 2; ++mt) {
                v2f av;
                av.x = w5c[(mt * 16 + m) * 32 + k0];
                av.y = w5c[(mt * 16 + m) * 32 + k0 + 1];
                a5[mt] = wmma_f32_4(av, bv, a5[mt]);
            }
        }
#pragma unroll
        for (int mt = 0; mt < 2; ++mt)
#pragma unroll
            for (int r = 0; r < 8; ++r) {
                int ch = mt * 16 + r + hi;
                att[2 + mt][r] = (a5[mt][r] + b5c[ch]) * tile[512 + ch * 16 + n];
            }
    }
    __syncthreads();

    // ----- h = xp * att, then out = w_pl @ h + b_pl; *scale + shortcut -----
    load_tile(tile, y + ((size_t)b * 128 + 64) * SP + s0, 64, lane);  // xp tile
    __syncthreads();
#pragma unroll
    for (int mt = 0; mt < 4; ++mt)
#pragma unroll
        for (int r = 0; r < 8; ++r) {
            int ch = mt * 16 + r + hi;
            tile[ch * 16 + n] = tile[ch * 16 + n] * att[mt][r];
        }
    __syncthreads();

    v8f ao[4] = {};
    for (int kk = 0; kk < 64; kk += 4) {
        int k0 = kk + khalf;
        v2f bv; bv.x = tile[k0 * 16 + n]; bv.y = tile[(k0 + 1) * 16 + n];
#pragma unroll
        for (int mt = 0; mt < 4; ++mt) {
            v2f av;
            av.x = wpl[(mt * 16 + m) * 64 + k0];
            av.y = wpl[(mt * 16 + m) * 64 + k0 + 1];
            ao[mt] = wmma_f32_4(av, bv, ao[mt]);
        }
    }

    const float* xb = x + (size_t)b * 64 * SP + s0;
    float* ob = out + (size_t)b * 64 * SP + s0;
#pragma unroll
    for (int mt = 0; mt < 4; ++mt)
#pragma unroll
        for (int r = 0; r < 8; ++r) {
            int ch = mt * 16 + r + hi;
            ob[(size_t)ch * SP + n] =
                (ao[mt][r] + bpl[ch]) * scl[ch] + xb[(size_t)ch * SP + n];
        }
}

// ---------------------------------------------------------------------------
extern "C" void kernel_launch(void* const* d_in, const int* in_sizes, int n_in,
                              void* d_out, int out_size, void* d_ws, size_t ws_size,
                              hipStream_t stream) {
    (void)in_sizes; (void)n_in; (void)out_size; (void)ws_size;

    const float* x    = (const float*)d_in[0];
    const float* ln_w = (const float*)d_in[1];
    const float* ln_b = (const float*)d_in[2];
    const float* w_pf = (const float*)d_in[3];
    const float* b_pf = (const float*)d_in[4];
    const float* w3a  = (const float*)d_in[5];
    const float* b3a  = (const float*)d_in[6];
    const float* w3b  = (const float*)d_in[7];
    const float* b3b  = (const float*)d_in[8];
    const float* w3c  = (const float*)d_in[9];
    const float* b3c  = (const float*)d_in[10];
    const float* wx3  = (const float*)d_in[11];
    const float* bx3  = (const float*)d_in[12];
    const float* w5a  = (const float*)d_in[13];
    const float* b5a  = (const float*)d_in[14];
    const float* w5b  = (const float*)d_in[15];
    const float* b5b  = (const float*)d_in[16];
    const float* w5c  = (const float*)d_in[17];
    const float* b5c  = (const float*)d_in[18];
    const float* wx5  = (const float*)d_in[19];
    const float* bx5  = (const float*)d_in[20];
    const float* w_pl = (const float*)d_in[21];
    const float* b_pl = (const float*)d_in[22];
    const float* scl  = (const float*)d_in[23];
    float* out = (float*)d_out;

    // workspace layout (floats)
    float* ws  = (float*)d_ws;
    float* y   = ws;                               // 2*128*SP = 33,554,432
    float* u3  = y  + (size_t)NB * 128 * SP;       // 2*32*SP each below
    float* g3  = u3 + (size_t)NB * 32 * SP;
    float* u5  = g3 + (size_t)NB * 32 * SP;
    float* g5  = u5 + (size_t)NB * 32 * SP;
    float* tmp = g5 + (size_t)NB * 32 * SP;

    const int grpBlocks = (NB * SP / 16) / 4;      // 16384 pos-groups / 4 waves = 4096
    const int dwBlocks  = NB * 32 * 32 * (64 / 8) * (64 / 32);  // 32768 tiles

    // 1) LayerNorm + pointwise 64->128 (WMMA)
    pf_ln_wmma<<<grpBlocks, 128, 0, stream>>>(x, ln_w, ln_b, w_pf, b_pf, y);

    // 2) depthwise chains, LDS-tiled (pads = dil*(K/2), matching the reference)
    dw_conv_tiled<3, 1, 8, 32><<<dwBlocks, 256, 0, stream>>>(y,   128,  0, w3a, b3a, tmp);  // a1 -> tmp
    dw_conv_tiled<5, 2, 8, 32><<<dwBlocks, 256, 0, stream>>>(tmp,  32,  0, w3b, b3b, u3);   // tmp -> u3
    dw_conv_tiled<3, 1, 8, 32><<<dwBlocks, 256, 0, stream>>>(y,   128,  0, wx3, bx3, g3);   // a1 -> g3
    dw_conv_tiled<5, 1, 8, 32><<<dwBlocks, 256, 0, stream>>>(y,   128, 32, w5a, b5a, tmp);  // a2 -> tmp
    dw_conv_tiled<7, 3, 8, 32><<<dwBlocks, 256, 0, stream>>>(tmp,  32,  0, w5b, b5b, u5);   // tmp -> u5
    dw_conv_tiled<5, 1, 8, 32><<<dwBlocks, 256, 0, stream>>>(y,   128, 32, wx5, bx5, g5);   // a2 -> g5

    // 3) gating + pointwise GEMMs + residual (WMMA)
    combine_wmma<<<grpBlocks, 128, 0, stream>>>(y, u3, g3, u5, g5,
                                                w3c, b3c, w5c, b5c,
                                                w_pl, b_pl, scl, x, out);
}